// HashEncoding_46909632807503
// MI455X (gfx1250) — compile-verified
//
#include <hip/hip_runtime.h>
#include <cstdint>
#include <cmath>

// Instant-NGP multiresolution hash encoding for MI455X (gfx1250).
// Key idea: per-level embedding tables (<=128KB) are staged into LDS with the
// CDNA5 Tensor Data Mover (tensor_load_to_lds, tracked by TENSORcnt), double
// buffered so DMA of level l+1 overlaps interpolation of level l. All corner
// gathers then hit LDS (ds_load_b64) instead of random VMEM.

#define NUM_LEVELS 16
#define TABLE_MAX 16384            // max entries per level
#define FEAT 2
#define BLOCK_THREADS 256
#define POINTS_PER_THREAD 8
#define POINTS_PER_BLOCK (BLOCK_THREADS * POINTS_PER_THREAD)
#define BUF_FLOATS (TABLE_MAX * FEAT)   // 32768 floats = 128KB per buffer

struct LevelParams {
  int res[NUM_LEVELS];
  int count[NUM_LEVELS];
  int offset[NUM_LEVELS];   // row offset into embeddings
  int dense[NUM_LEVELS];    // 1 = dense grid indexing, 0 = spatial hash
};

typedef unsigned int v4u __attribute__((ext_vector_type(4)));
typedef int v4i __attribute__((ext_vector_type(4)));
typedef int v8i __attribute__((ext_vector_type(8)));

#if defined(__has_builtin)
#if __has_builtin(__builtin_amdgcn_tensor_load_to_lds)
#define HAS_TDM 1
#endif
#endif

// Stage one level's table (nfloat 4-byte elements) from global to LDS.
// Called by one wave; TDM ignores EXEC and issues once per wave.
__device__ __forceinline__ void stage_level(const float* __restrict__ src,
                                            float* dst, uint32_t nfloat) {
#if defined(HAS_TDM)
  // D# per CDNA5 ISA.
  uint64_t ga = (uint64_t)(uintptr_t)src;
  uint32_t lds_off = (uint32_t)(uintptr_t)dst;  // low 32 bits = LDS byte offset
  v4u g0;
  g0[0] = 1u;                                        // count=1, user descriptor
  g0[1] = lds_off;                                   // lds_addr (bytes)
  g0[2] = (uint32_t)ga;                              // global_addr[31:0]
  g0[3] = (uint32_t)((ga >> 32) & 0x01ffffffu)       // global_addr[56:32]
          | (2u << 30);                              // type = 2 ("image")
  v8i g1;
  g1[0] = (int)(2u << 16);                           // data_size=2 -> 4 bytes
  g1[1] = (int)((nfloat & 0xffffu) << 16);           // tensor_dim0[15:0]
  g1[2] = (int)(((nfloat >> 16) & 0xffffu)           // tensor_dim0[31:16]
                | (1u << 16));                       // tensor_dim1 = 1
  g1[3] = (int)((nfloat & 0xffffu) << 16);           // tile_dim0 = nfloat (<=32768)
  g1[4] = 1;                                         // tile_dim1=1, tile_dim2=0
  g1[5] = (int)nfloat;                               // tensor_dim0_stride
  g1[6] = 0;                                         // stride hi / dim1_stride lo
  g1[7] = 0;
  v4i g2;  g2[0] = 1; g2[1] = 1; g2[2] = 0; g2[3] = 0;   // tensor_dim2/3 = 1
  v4i g3;  g3[0] = 0; g3[1] = (int)(1u << 16); g3[2] = 0; g3[3] = 0; // dim4=1
#if __clang_major__ >= 23
  v8i g4;  g4[0]=0; g4[1]=0; g4[2]=0; g4[3]=0; g4[4]=0; g4[5]=0; g4[6]=0; g4[7]=0;
  __builtin_amdgcn_tensor_load_to_lds(g0, g1, g2, g3, g4, 0);
#else
  __builtin_amdgcn_tensor_load_to_lds(g0, g1, g2, g3, 0);
#endif
#else
  // Fallback: cooperative copy by the calling wave.
  uint32_t lane = threadIdx.x & 31u;
  for (uint32_t i = lane; i < nfloat; i += 32u) dst[i] = src[i];
#endif
}

template <bool DENSE>
__device__ __forceinline__ void do_level(const float* __restrict__ buf, int res,
                                         const float* px, const float* py,
                                         const float* pz,
                                         float* __restrict__ out, int pbase,
                                         int tid, int l, int npoints) {
  const float fres = (float)res;
  const uint32_t r1 = (uint32_t)(res + 1);
#pragma unroll
  for (int k = 0; k < POINTS_PER_THREAD; ++k) {
    int p = pbase + k * BLOCK_THREADS + tid;
    if (p >= npoints) continue;
    float sx = px[k] * fres, sy = py[k] * fres, sz = pz[k] * fres;
    float fx = floorf(sx), fy = floorf(sy), fz = floorf(sz);
    float tx = sx - fx, ty = sy - fy, tz = sz - fz;
    int ix = (int)fx, iy = (int)fy, iz = (int)fz;
    uint32_t x0 = (uint32_t)min(ix, res),     x1 = (uint32_t)min(ix + 1, res);
    uint32_t y0 = (uint32_t)min(iy, res),     y1 = (uint32_t)min(iy + 1, res);
    uint32_t z0 = (uint32_t)min(iz, res),     z1 = (uint32_t)min(iz + 1, res);
    uint32_t hx0, hx1, hy0, hy1, hz0, hz1;
    if (DENSE) {
      hx0 = x0;            hx1 = x1;
      hy0 = y0 * r1;       hy1 = y1 * r1;
      hz0 = z0 * r1 * r1;  hz1 = z1 * r1 * r1;
    } else {
      // low 14 bits of (c*prime) are exact in 32-bit arithmetic; %16384 on a
      // power of two == &16383 for numpy signed remainder semantics too.
      hx0 = x0;                     hx1 = x1;
      hy0 = y0 * 2654435761u;       hy1 = y1 * 2654435761u;
      hz0 = z0 * 805459861u;        hz1 = z1 * 805459861u;
    }
    float wx0 = 1.f - tx, wx1 = tx;
    float wy0 = 1.f - ty, wy1 = ty;
    float wz0 = 1.f - tz, wz1 = tz;
    float a0 = 0.f, a1 = 0.f;
#define CORN(HX, HY, HZ, WX, WY, WZ)                                       \
  {                                                                        \
    uint32_t idx_ = DENSE ? (HX + HY + HZ)                                 \
                          : ((HX ^ HY ^ HZ) & (uint32_t)(TABLE_MAX - 1));  \
    const float2 e_ = *(const float2*)(buf + 2u * idx_);                   \
    float w_ = (WX) * (WY) * (WZ);                                         \
    a0 = fmaf(w_, e_.x, a0);                                               \
    a1 = fmaf(w_, e_.y, a1);                                               \
  }
    // Reference corner order: x slowest, z fastest (meshgrid 'ij').
    CORN(hx0, hy0, hz0, wx0, wy0, wz0)
    CORN(hx0, hy0, hz1, wx0, wy0, wz1)
    CORN(hx0, hy1, hz0, wx0, wy1, wz0)
    CORN(hx0, hy1, hz1, wx0, wy1, wz1)
    CORN(hx1, hy0, hz0, wx1, wy0, wz0)
    CORN(hx1, hy0, hz1, wx1, wy0, wz1)
    CORN(hx1, hy1, hz0, wx1, wy1, wz0)
    CORN(hx1, hy1, hz1, wx1, wy1, wz1)
#undef CORN
    float2 r;
    r.x = a0;
    r.y = a1;
    *(float2*)(out + (size_t)p * (NUM_LEVELS * FEAT) + 2 * l) = r;
  }
}

__global__ __launch_bounds__(BLOCK_THREADS) void hashenc_tdm_kernel(
    const float* __restrict__ coords, const float* __restrict__ emb,
    float* __restrict__ out, int npoints, LevelParams lp) {
  extern __shared__ float lds_tab[];  // 2 buffers x 32768 floats = 256KB
  const int tid = (int)threadIdx.x;
  const int pbase = (int)blockIdx.x * POINTS_PER_BLOCK;

  float px[POINTS_PER_THREAD], py[POINTS_PER_THREAD], pz[POINTS_PER_THREAD];
#pragma unroll
  for (int k = 0; k < POINTS_PER_THREAD; ++k) {
    int p = pbase + k * BLOCK_THREADS + tid;
    if (p < npoints) {
      px[k] = coords[3 * (size_t)p + 0];
      py[k] = coords[3 * (size_t)p + 1];
      pz[k] = coords[3 * (size_t)p + 2];
    } else {
      px[k] = py[k] = pz[k] = 0.f;
    }
  }

  // Prime the pipeline: DMA level 0 into buffer 0.
  if (tid < 32) {
    stage_level(emb + 2 * (size_t)lp.offset[0], lds_tab,
                (uint32_t)lp.count[0] * 2u);
  }

#pragma unroll 1
  for (int l = 0; l < NUM_LEVELS; ++l) {
    float* buf = lds_tab + (size_t)(l & 1) * BUF_FLOATS;
    if (l + 1 < NUM_LEVELS) {
      // Kick off DMA for level l+1 into the other buffer, then wait until at
      // most that one transfer is outstanding -> level l's data has landed.
      if (tid < 32) {
        stage_level(emb + 2 * (size_t)lp.offset[l + 1],
                    lds_tab + (size_t)((l + 1) & 1) * BUF_FLOATS,
                    (uint32_t)lp.count[l + 1] * 2u);
      }
#if defined(HAS_TDM)
      __builtin_amdgcn_s_wait_tensorcnt(1);
#endif
    } else {
#if defined(HAS_TDM)
      __builtin_amdgcn_s_wait_tensorcnt(0);
#endif
    }
    __syncthreads();  // publish LDS table to all waves

    if (lp.dense[l]) {
      do_level<true>(buf, lp.res[l], px, py, pz, out, pbase, tid, l, npoints);
    } else {
      do_level<false>(buf, lp.res[l], px, py, pz, out, pbase, tid, l, npoints);
    }

    __syncthreads();  // all reads of buf done before it is overwritten (l+2)
  }
}

extern "C" void kernel_launch(void* const* d_in, const int* in_sizes, int n_in,
                              void* d_out, int out_size, void* d_ws,
                              size_t ws_size, hipStream_t stream) {
  (void)n_in; (void)out_size; (void)d_ws; (void)ws_size;
  const float* coords = (const float*)d_in[0];
  const float* emb = (const float*)d_in[1];
  float* out = (float*)d_out;
  const int npoints = in_sizes[0] / 3;

  // Replicate numpy's level-resolution math exactly (double exp/log/pow/floor).
  LevelParams lp;
  const double b = exp((log(512.0) - log(16.0)) / (double)(NUM_LEVELS - 1));
  long long off = 0;
  for (int l = 0; l < NUM_LEVELS; ++l) {
    int res = (int)floor(16.0 * pow(b, (double)l));
    long long densec = (long long)(res + 1) * (res + 1) * (res + 1);
    long long cnt = densec < (long long)TABLE_MAX ? densec : (long long)TABLE_MAX;
    lp.res[l] = res;
    lp.count[l] = (int)cnt;
    lp.offset[l] = (int)off;
    lp.dense[l] = (densec == cnt) ? 1 : 0;
    off += cnt;
  }

  const int blocks = (npoints + POINTS_PER_BLOCK - 1) / POINTS_PER_BLOCK;
  const size_t shmem = (size_t)2 * BUF_FLOATS * sizeof(float);  // 256 KB
  (void)hipFuncSetAttribute((const void*)hashenc_tdm_kernel,
                            hipFuncAttributeMaxDynamicSharedMemorySize,
                            (int)shmem);
  hashenc_tdm_kernel<<<blocks, BLOCK_THREADS, shmem, stream>>>(coords, emb, out,
                                                               npoints, lp);
}